// FlashFFTConvInterface_81716047773932
// MI455X (gfx1250) — compile-verified
//
#include <hip/hip_runtime.h>
#include <hip/hip_bf16.h>

typedef float v2f __attribute__((ext_vector_type(2)));
typedef float v8f __attribute__((ext_vector_type(8)));

#define TWO_PI_256 0.0245436926f   // 2*pi/256
#define TWO_PI_16  0.3926990817f   // 2*pi/16
#define PADW 133                   // odd row stride -> conflict-free column FFTs

// ---------------------------------------------------------------------------
// CDNA5 async data movers (ASYNCcnt-tracked HBM <-> LDS, 16B per lane).
// INST_OFFSET is added to BOTH the LDS and global address (ISA 15.18.3),
// so one address setup covers a whole 1KB row with two instructions.
// ASYNC loads complete in order, and stores in order (ISA 08 §4.1), which
// makes partial waits (s_wait_asynccnt 2) a valid double-buffer drain.
// ---------------------------------------------------------------------------
__device__ __forceinline__ void async_load_row(const float* gsrc, unsigned ldsByteOff, int lane)
{
    unsigned long long ga = (unsigned long long)(uintptr_t)gsrc + (unsigned)lane * 16u;
    unsigned la = ldsByteOff + (unsigned)lane * 16u;
    asm volatile("global_load_async_to_lds_b128 %0, %1, off\n\t"
                 "global_load_async_to_lds_b128 %0, %1, off offset:512"
                 :: "v"(la), "v"(ga) : "memory");
}
__device__ __forceinline__ void async_store_row(float* gdst, unsigned ldsByteOff, int lane)
{
    unsigned long long ga = (unsigned long long)(uintptr_t)gdst + (unsigned)lane * 16u;
    unsigned la = ldsByteOff + (unsigned)lane * 16u;
    asm volatile("global_store_async_from_lds_b128 %1, %0, off\n\t"
                 "global_store_async_from_lds_b128 %1, %0, off offset:512"
                 :: "v"(la), "v"(ga) : "memory");
}
__device__ __forceinline__ void wait_async0() { asm volatile("s_wait_asynccnt 0" ::: "memory"); }
__device__ __forceinline__ void wait_async2() { asm volatile("s_wait_asynccnt 2" ::: "memory"); }
__device__ __forceinline__ void wait_ds()     { asm volatile("s_wait_dscnt 0"    ::: "memory"); }

// ---------------------------------------------------------------------------
// One 256-point FFT executed cooperatively by ONE wave, radix-16 Cooley-Tukey:
// two 16x16 complex matmuls + elementwise 16x16 twiddle (table-driven).
// Each 16x16x16 real matmul = 4 chained v_wmma_f32_16x16x4_f32.
// REAL_IN=true skips the Ai terms in stage 1 (8 WMMAs instead of 16).
// sgn = -1 forward, +1 inverse. In-place on (re, im) with element stride.
// ---------------------------------------------------------------------------
template<bool REAL_IN>
__device__ __forceinline__ void fft256_wave(float* re, float* im, int stride,
                                            const float* Fc, const float* Fs,
                                            const float* Tc, const float* Ts,
                                            float sgn, float scale,
                                            float* spR, float* spI, int lane)
{
    const int mA   = lane & 15;          // A-frag row (M)
    const int nC   = lane & 15;          // B/C/D-frag col (N)
    const int kHi  = (lane >> 4) << 1;   // K sub-offset: lanes 0-15 -> {0,1}, 16-31 -> {2,3}
    const int mOff = (lane >> 4) << 3;   // C/D-frag M offset: 0 or 8

    // ---- stage 1: B = A * F,  A[m][k] = x[16k + m], F = Fc + i*sgn*Fs
    v8f br = {}, bi = {};
#pragma unroll
    for (int kc = 0; kc < 4; ++kc) {
        const int k0 = kc * 4 + kHi;
        v2f ar, fc, fsp;
        ar.x = re[(16 * k0 + mA) * stride];
        ar.y = re[(16 * (k0 + 1) + mA) * stride];
        fc.x = Fc[k0 * 16 + nC];
        fc.y = Fc[(k0 + 1) * 16 + nC];
        const float s0 = Fs[k0 * 16 + nC] * sgn;
        const float s1 = Fs[(k0 + 1) * 16 + nC] * sgn;
        fsp.x = s0;  fsp.y = s1;
        br = __builtin_amdgcn_wmma_f32_16x16x4_f32(false, ar, false, fc,  (short)0, br, false, false);
        bi = __builtin_amdgcn_wmma_f32_16x16x4_f32(false, ar, false, fsp, (short)0, bi, false, false);
        if constexpr (!REAL_IN) {
            v2f ai, fsn;
            ai.x = im[(16 * k0 + mA) * stride];
            ai.y = im[(16 * (k0 + 1) + mA) * stride];
            fsn.x = -s0; fsn.y = -s1;
            br = __builtin_amdgcn_wmma_f32_16x16x4_f32(false, ai, false, fsn, (short)0, br, false, false);
            bi = __builtin_amdgcn_wmma_f32_16x16x4_f32(false, ai, false, fc,  (short)0, bi, false, false);
        }
    }

    // ---- twiddle W256^{m*n} (precomputed table), spill 16x16 to LDS
#pragma unroll
    for (int r = 0; r < 8; ++r) {
        const int m = r + mOff;
        const float c = Tc[m * 16 + nC];
        const float s = Ts[m * 16 + nC] * sgn;
        const float x = br[r], y = bi[r];
        spR[m * 16 + nC] = x * c - y * s;
        spI[m * 16 + nC] = x * s + y * c;
    }

    // ---- stage 2: D = F * C (complex)
    v8f dr = {}, di = {};
#pragma unroll
    for (int kc = 0; kc < 4; ++kc) {
        const int k0 = kc * 4 + kHi;
        v2f fac, fasp, fasn, cr, ci;
        fac.x = Fc[mA * 16 + k0];
        fac.y = Fc[mA * 16 + k0 + 1];
        const float s0 = Fs[mA * 16 + k0] * sgn;
        const float s1 = Fs[mA * 16 + k0 + 1] * sgn;
        fasp.x = s0;  fasp.y = s1;
        fasn.x = -s0; fasn.y = -s1;
        cr.x = spR[k0 * 16 + nC];
        cr.y = spR[(k0 + 1) * 16 + nC];
        ci.x = spI[k0 * 16 + nC];
        ci.y = spI[(k0 + 1) * 16 + nC];
        dr = __builtin_amdgcn_wmma_f32_16x16x4_f32(false, fac,  false, cr, (short)0, dr, false, false);
        dr = __builtin_amdgcn_wmma_f32_16x16x4_f32(false, fasn, false, ci, (short)0, dr, false, false);
        di = __builtin_amdgcn_wmma_f32_16x16x4_f32(false, fac,  false, ci, (short)0, di, false, false);
        di = __builtin_amdgcn_wmma_f32_16x16x4_f32(false, fasp, false, cr, (short)0, di, false, false);
    }

    // ---- output: element (M=k2, N=k1) -> X[k1 + 16*k2]
#pragma unroll
    for (int r = 0; r < 8; ++r) {
        const int m = r + mOff;
        const int k = nC + 16 * m;
        re[k * stride] = dr[r] * scale;
        im[k * stride] = di[r] * scale;
    }
}

// LDS layout (dynamic):
//   Fc/Fs[256] : 16-pt DFT tables     Tc/Ts[256] : 256-pt twiddle tables
//   Sr/Si[256*PADW] : Hermitian-packed plane spectrum (129 cols kept, pad->133)
//   per wave (8): rowR0[256], rowR1[256] (double buffer), rowI[256], spR[256], spI[256]
#define WAVE_FLOATS 1280
#define SMEM_FLOATS (4 * 256 + 2 * 256 * PADW + 8 * WAVE_FLOATS)   // 317,440 B

#define SETUP_LDS()                                                     \
    extern __shared__ float smem[];                                     \
    float* Fc = smem;                                                   \
    float* Fs = Fc + 256;                                               \
    float* Tc = Fs + 256;                                               \
    float* Ts = Tc + 256;                                               \
    float* Sr = Ts + 256;                                               \
    float* Si = Sr + 256 * PADW;                                        \
    float* wbase = Si + 256 * PADW;                                     \
    const int tid = threadIdx.x;                                        \
    const int lane = tid & 31;                                          \
    const int wave = tid >> 5;                                          \
    float* rowR0 = wbase + wave * WAVE_FLOATS;                          \
    float* rowR1 = rowR0 + 256;                                         \
    float* rowI  = rowR1 + 256;                                         \
    float* spR   = rowI + 256;                                          \
    float* spI   = spR + 256;                                           \
    const unsigned off0 = (unsigned)(4 * 256 + 2 * 256 * PADW + wave * WAVE_FLOATS) * 4u; \
    const unsigned off1 = off0 + 1024u;                                 \
    (void)off0; (void)off1;                                             \
    {   const int kk = tid >> 4, nn = tid & 15;                         \
        const float a16  = TWO_PI_16  * (float)(kk * nn);               \
        const float a256 = TWO_PI_256 * (float)(kk * nn);               \
        Fc[tid] = __cosf(a16);  Fs[tid] = __sinf(a16);                  \
        Tc[tid] = __cosf(a256); Ts[tid] = __sinf(a256); }               \
    __syncthreads();

// Double-buffered forward row FFT pass: stream rows HBM->LDS with the async
// engine one iteration ahead of the WMMA FFT consuming them.
#define FWD_ROWS_REAL(plane)                                            \
    async_load_row((plane) + wave * 256, off0, lane);                   \
    {   int it = 0;                                                     \
        for (int r = wave; r < 256; r += 8, ++it) {                     \
            float* curR = (it & 1) ? rowR1 : rowR0;                     \
            const unsigned nxtOff = (it & 1) ? off0 : off1;             \
            if (r + 8 < 256) { async_load_row((plane) + (r + 8) * 256, nxtOff, lane); wait_async2(); } \
            else             { wait_async0(); }                         \
            fft256_wave<true>(curR, rowI, 1, Fc, Fs, Tc, Ts, -1.f, 1.f, spR, spI, lane); \
            for (int kk = lane; kk < 129; kk += 32) {                   \
                Sr[r * PADW + kk] = curR[kk];                           \
                Si[r * PADW + kk] = rowI[kk];                           \
            }                                                           \
        }                                                               \
    }

// ---------------------------------------------------------------------------
// Kernel 1: k_f = rfft2(k) for all 64 channels -> ws as float2[64][256][129]
// ---------------------------------------------------------------------------
__global__ void __launch_bounds__(256) kf_build_kernel(const float* __restrict__ kin,
                                                       float2* __restrict__ kf)
{
    SETUP_LDS();
    const int ch = blockIdx.x;
    const float* kplane = kin + (size_t)ch * 65536;

    FWD_ROWS_REAL(kplane);
    __syncthreads();

    for (int c = wave; c < 129; c += 8)
        fft256_wave<false>(Sr + c, Si + c, PADW, Fc, Fs, Tc, Ts, -1.f, 1.f, spR, spI, lane);
    __syncthreads();

    float2* out = kf + (size_t)ch * 33024;  // 256*129
    for (int idx = tid; idx < 33024; idx += 256) {
        const int i = idx / 129;
        const int c = idx - i * 129;
        out[idx] = make_float2(Sr[i * PADW + c], Si[i * PADW + c]);
    }
}

// ---------------------------------------------------------------------------
// Kernel 2: fused per-plane  y = irfft2(rfft2(u) * k_f).  One WG per (b,c).
// Plane spectrum stays resident in 320KB LDS: u read once, y written once.
// ---------------------------------------------------------------------------
__global__ void __launch_bounds__(256) fftconv_plane_kernel(const float* __restrict__ u,
                                                            const float2* __restrict__ kf,
                                                            float* __restrict__ y, int C)
{
    SETUP_LDS();
    const int bc = blockIdx.x;
    const int ch = bc % C;
    const float* uplane = u + (size_t)bc * 65536;

    // ---- forward row FFTs (real input, pipelined async HBM->LDS staging)
    FWD_ROWS_REAL(uplane);
    __syncthreads();

    // ---- per column: forward FFT, pointwise * k_f, inverse FFT (fused)
    const float2* kfc = kf + (size_t)ch * 33024;
    for (int c = wave; c < 129; c += 8) {
        fft256_wave<false>(Sr + c, Si + c, PADW, Fc, Fs, Tc, Ts, -1.f, 1.f, spR, spI, lane);
        for (int i = lane; i < 256; i += 32) {
            const float2 w = kfc[i * 129 + c];
            const float a = Sr[i * PADW + c];
            const float b = Si[i * PADW + c];
            Sr[i * PADW + c] = a * w.x - b * w.y;
            Si[i * PADW + c] = a * w.y + b * w.x;
        }
        fft256_wave<false>(Sr + c, Si + c, PADW, Fc, Fs, Tc, Ts, 1.f, 1.f / 256.f, spR, spI, lane);
    }
    __syncthreads();

    // ---- inverse row FFTs with Hermitian reconstruction; async LDS->HBM
    //      stores drained lazily (two iterations deep, in-order completion)
    float* yplane = y + (size_t)bc * 65536;
    {
        int it = 0;
        for (int r = wave; r < 256; r += 8, ++it) {
            float* curR = (it & 1) ? rowR1 : rowR0;
            const unsigned curOff = (it & 1) ? off1 : off0;
            wait_async2();   // store issued 2 iters ago (same buffer) has finished reading LDS
            for (int n = lane; n < 256; n += 32) {
                const int idx = (n <= 128) ? n : 256 - n;
                curR[n] = Sr[r * PADW + idx];
                rowI[n] = (n <= 128) ? Si[r * PADW + idx] : -Si[r * PADW + idx];
            }
            fft256_wave<false>(curR, rowI, 1, Fc, Fs, Tc, Ts, 1.f, 1.f / 256.f, spR, spI, lane);
            wait_ds();       // fft's ds_stores visible in LDS before the async engine reads it
            async_store_row(yplane + r * 256, curOff, lane);
        }
    }
    wait_async0();
}

extern "C" void kernel_launch(void* const* d_in, const int* in_sizes, int n_in,
                              void* d_out, int out_size, void* d_ws, size_t ws_size,
                              hipStream_t stream) {
    (void)in_sizes; (void)n_in; (void)out_size; (void)ws_size;
    const float* u = (const float*)d_in[0];   // [8,64,256,256]
    const float* k = (const float*)d_in[1];   // [64,256,256]
    float* y   = (float*)d_out;               // [8,64,256,256]
    float2* kf = (float2*)d_ws;               // [64,256,129] complex, 16.9 MB

    const size_t smem = (size_t)SMEM_FLOATS * sizeof(float); // 317,440 B <= 320 KB/WGP
    hipFuncSetAttribute((const void*)kf_build_kernel,
                        hipFuncAttributeMaxDynamicSharedMemorySize, (int)smem);
    hipFuncSetAttribute((const void*)fftconv_plane_kernel,
                        hipFuncAttributeMaxDynamicSharedMemorySize, (int)smem);

    kf_build_kernel<<<64, 256, smem, stream>>>(k, kf);
    fftconv_plane_kernel<<<512, 256, smem, stream>>>(u, kf, y, 64);
}